// Face3D_5995774345512
// MI455X (gfx1250) — compile-verified
//
#include <hip/hip_runtime.h>
#include <math.h>

#define N_VERT  35709
#define N_FACE  70789
#define BATCH   32
#define M3      (3 * N_VERT)          // 107127 rows
#define NTILES  ((M3 + 15) / 16)      // 6696 M-tiles of 16

typedef float v2f __attribute__((ext_vector_type(2)));
typedef float v8f __attribute__((ext_vector_type(8)));

// ---------------------------------------------------------------------------
// Kernel 0: meanshape centroid (3 floats) + per-batch rotation matrices R=Rz*Ry*Rx
// ---------------------------------------------------------------------------
__global__ __launch_bounds__(256) void k_prep(const float* __restrict__ coeff,
                                              const float* __restrict__ meanshape,
                                              float* __restrict__ center,
                                              float* __restrict__ Rws) {
    __shared__ float red[3][256];
    int t = threadIdx.x;
    float s0 = 0.f, s1 = 0.f, s2 = 0.f;
    for (int v = t; v < N_VERT; v += 256) {
        s0 += meanshape[3 * v + 0];
        s1 += meanshape[3 * v + 1];
        s2 += meanshape[3 * v + 2];
    }
    red[0][t] = s0; red[1][t] = s1; red[2][t] = s2;
    __syncthreads();
    for (int off = 128; off > 0; off >>= 1) {
        if (t < off) {
            red[0][t] += red[0][t + off];
            red[1][t] += red[1][t + off];
            red[2][t] += red[2][t + off];
        }
        __syncthreads();
    }
    if (t == 0) {
        center[0] = red[0][0] / (float)N_VERT;
        center[1] = red[1][0] / (float)N_VERT;
        center[2] = red[2][0] / (float)N_VERT;
    }
    if (t < BATCH) {
        float ax = coeff[t * 277 + 244];
        float ay = coeff[t * 277 + 245];
        float az = coeff[t * 277 + 246];
        float cx = cosf(ax), sx = sinf(ax);
        float cy = cosf(ay), sy = sinf(ay);
        float cz = cosf(az), sz = sinf(az);
        float* R = Rws + t * 9;           // row-major R = Rz*Ry*Rx
        R[0] = cz * cy;
        R[1] = -sz * cx + cz * sy * sx;
        R[2] = sz * sx + cz * sy * cx;
        R[3] = sz * cy;
        R[4] = cz * cx + sz * sy * sx;
        R[5] = -cz * sx + sz * sy * cx;
        R[6] = -sy;
        R[7] = cy * sx;
        R[8] = cy * cx;
    }
}

// ---------------------------------------------------------------------------
// Kernel A: fused FP32 WMMA GEMMs.
//   fsws[row*32+b]  = (idBase@id_c + exBase@ex_c)[row,b] + meanshape[row] - center[row%3]
//   texws[row*32+b] = (texBase@tex_c)[row,b] + meantex[row]
// One wave per 16-row M tile; two 16-batch N tiles per wave.
// ---------------------------------------------------------------------------
__global__ __launch_bounds__(256) void k_gemm(const float* __restrict__ idBase,
                                              const float* __restrict__ exBase,
                                              const float* __restrict__ texBase,
                                              const float* __restrict__ coeff,
                                              const float* __restrict__ meanshape,
                                              const float* __restrict__ meantex,
                                              const float* __restrict__ center,
                                              float* __restrict__ fsws,
                                              float* __restrict__ texws) {
    int wave = threadIdx.x >> 5;
    int lane = threadIdx.x & 31;
    int tile = blockIdx.x * 8 + wave;
    if (tile >= NTILES) return;                    // wave-uniform exit
    int m0   = tile * 16;
    int half = lane >> 4;                          // 0 or 1
    int l15  = lane & 15;
    int kOff = 2 * half;
    int rowA = m0 + l15; if (rowA > M3 - 1) rowA = M3 - 1;

    v8f accF0 = {}, accF1 = {}, accT0 = {}, accT1 = {};

    // --- idBase, K = 80 ---
    for (int kb = 0; kb < 80; kb += 4) {
        v2f a = *(const v2f*)(idBase + (size_t)rowA * 80 + kb + kOff);
        int kc = kb + kOff;
        v2f b0, b1;
        b0.x = coeff[l15 * 277 + kc];            b0.y = coeff[l15 * 277 + kc + 1];
        b1.x = coeff[(l15 + 16) * 277 + kc];     b1.y = coeff[(l15 + 16) * 277 + kc + 1];
        accF0 = __builtin_amdgcn_wmma_f32_16x16x4_f32(false, a, false, b0, (short)0, accF0, false, false);
        accF1 = __builtin_amdgcn_wmma_f32_16x16x4_f32(false, a, false, b1, (short)0, accF1, false, false);
    }
    // --- exBase, K = 64 (coeff cols 80..143) ---
    for (int kb = 0; kb < 64; kb += 4) {
        v2f a = *(const v2f*)(exBase + (size_t)rowA * 64 + kb + kOff);
        int kc = 80 + kb + kOff;
        v2f b0, b1;
        b0.x = coeff[l15 * 277 + kc];            b0.y = coeff[l15 * 277 + kc + 1];
        b1.x = coeff[(l15 + 16) * 277 + kc];     b1.y = coeff[(l15 + 16) * 277 + kc + 1];
        accF0 = __builtin_amdgcn_wmma_f32_16x16x4_f32(false, a, false, b0, (short)0, accF0, false, false);
        accF1 = __builtin_amdgcn_wmma_f32_16x16x4_f32(false, a, false, b1, (short)0, accF1, false, false);
    }
    // --- texBase, K = 100 (coeff cols 144..243) ---
    for (int kb = 0; kb < 100; kb += 4) {
        v2f a = *(const v2f*)(texBase + (size_t)rowA * 100 + kb + kOff);
        int kc = 144 + kb + kOff;
        v2f b0, b1;
        b0.x = coeff[l15 * 277 + kc];            b0.y = coeff[l15 * 277 + kc + 1];
        b1.x = coeff[(l15 + 16) * 277 + kc];     b1.y = coeff[(l15 + 16) * 277 + kc + 1];
        accT0 = __builtin_amdgcn_wmma_f32_16x16x4_f32(false, a, false, b0, (short)0, accT0, false, false);
        accT1 = __builtin_amdgcn_wmma_f32_16x16x4_f32(false, a, false, b1, (short)0, accT1, false, false);
    }

    float c0 = center[0], c1 = center[1], c2 = center[2];
    // D layout: VGPR r, lane l -> row = m0 + r + 8*(l>>4), batch = (l&15) [+16 for tile1]
    for (int r = 0; r < 8; ++r) {
        int row = m0 + r + 8 * half;
        if (row < M3) {
            float ms = meanshape[row];
            float mt = meantex[row];
            int   cm = row % 3;
            float cc = (cm == 0) ? c0 : ((cm == 1) ? c1 : c2);
            fsws[(size_t)row * 32 + l15]       = accF0[r] + ms - cc;
            fsws[(size_t)row * 32 + l15 + 16]  = accF1[r] + ms - cc;
            texws[(size_t)row * 32 + l15]      = accT0[r] + mt;
            texws[(size_t)row * 32 + l15 + 16] = accT1[r] + mt;
        }
    }
}

// ---------------------------------------------------------------------------
// Kernel B: per-face normals. lane = batch (coalesced 128B gathers).
// fnws has N_FACE+1 rows; last row is the appended zero row.
// ---------------------------------------------------------------------------
__global__ __launch_bounds__(256) void k_face(const float* __restrict__ fsws,
                                              const int* __restrict__ face_buf,
                                              float* __restrict__ fnws) {
    int lane = threadIdx.x;                     // batch
    int f = blockIdx.x * 8 + threadIdx.y;
    if (f > N_FACE) return;
    if (f == N_FACE) {
        fnws[((size_t)f * 3 + 0) * 32 + lane] = 0.f;
        fnws[((size_t)f * 3 + 1) * 32 + lane] = 0.f;
        fnws[((size_t)f * 3 + 2) * 32 + lane] = 0.f;
        return;
    }
    int i1 = face_buf[f * 3 + 0];
    int i2 = face_buf[f * 3 + 1];
    int i3 = face_buf[f * 3 + 2];
    float p1[3], p2[3], p3[3];
    #pragma unroll
    for (int c = 0; c < 3; ++c) {
        p1[c] = fsws[((size_t)i1 * 3 + c) * 32 + lane];
        p2[c] = fsws[((size_t)i2 * 3 + c) * 32 + lane];
        p3[c] = fsws[((size_t)i3 * 3 + c) * 32 + lane];
    }
    float ax = p1[0] - p2[0], ay = p1[1] - p2[1], az = p1[2] - p2[2];
    float bx = p2[0] - p3[0], by = p2[1] - p3[1], bz = p2[2] - p3[2];
    float nx = ay * bz - az * by;
    float ny = az * bx - ax * bz;
    float nz = ax * by - ay * bx;
    float nrm = fmaxf(sqrtf(nx * nx + ny * ny + nz * nz), 1e-12f);
    float inv = 1.f / nrm;
    fnws[((size_t)f * 3 + 0) * 32 + lane] = nx * inv;
    fnws[((size_t)f * 3 + 1) * 32 + lane] = ny * inv;
    fnws[((size_t)f * 3 + 2) * 32 + lane] = nz * inv;
}

// ---------------------------------------------------------------------------
// Kernel C: vertex normals -> rotate -> SH lighting -> face_color; fst = R*fs + t.
// lane = batch for coalesced gathers; LDS transpose for coalesced (B,N,3) stores.
// ---------------------------------------------------------------------------
__global__ __launch_bounds__(256) void k_vert(const float* __restrict__ fsws,
                                              const float* __restrict__ texws,
                                              const float* __restrict__ fnws,
                                              const float* __restrict__ Rws,
                                              const float* __restrict__ coeff,
                                              const int* __restrict__ point_buf,
                                              float* __restrict__ out_color,
                                              float* __restrict__ out_fst) {
    __shared__ float sR[BATCH * 9];
    __shared__ float sG[BATCH * 27];
    __shared__ float sT[BATCH * 3];
    __shared__ float sOut[8 * 32 * 6];

    int tid = threadIdx.y * 32 + threadIdx.x;
    for (int i = tid; i < BATCH * 9; i += 256) sR[i] = Rws[i];
    for (int i = tid; i < BATCH * 27; i += 256) {
        int b = i / 27, k = i % 27;
        sG[i] = coeff[b * 277 + 247 + k] + ((k % 9 == 0) ? 0.8f : 0.f);
    }
    for (int i = tid; i < BATCH * 3; i += 256) {
        int b = i / 3, j = i % 3;
        sT[i] = coeff[b * 277 + 274 + j];
    }
    __syncthreads();

    int lane = threadIdx.x;                    // batch
    int y = threadIdx.y;
    int v = blockIdx.x * 8 + y;
    if (v < N_VERT) {
        float sn0 = 0.f, sn1 = 0.f, sn2 = 0.f;
        #pragma unroll
        for (int j = 0; j < 8; ++j) {
            int fi = point_buf[v * 8 + j];
            sn0 += fnws[((size_t)fi * 3 + 0) * 32 + lane];
            sn1 += fnws[((size_t)fi * 3 + 1) * 32 + lane];
            sn2 += fnws[((size_t)fi * 3 + 2) * 32 + lane];
        }
        float inv = 1.f / fmaxf(sqrtf(sn0 * sn0 + sn1 * sn1 + sn2 * sn2), 1e-12f);
        sn0 *= inv; sn1 *= inv; sn2 *= inv;

        const float* R = sR + lane * 9;
        // norm_r = v_norm @ R^T  -> nr_c = sum_j R[c*3+j] * vn_j
        float nx = R[0] * sn0 + R[1] * sn1 + R[2] * sn2;
        float ny = R[3] * sn0 + R[4] * sn1 + R[5] * sn2;
        float nz = R[6] * sn0 + R[7] * sn1 + R[8] * sn2;

        float Y[9];
        Y[0] = 0.8862269255f;
        Y[1] = -1.7724538509f * ny;
        Y[2] =  1.7724538509f * nz;
        Y[3] = -1.7724538509f * nx;
        Y[4] =  2.4270509831f * nx * ny;
        Y[5] = -2.4270509831f * ny * nz;
        Y[6] =  0.7006292692f * (3.f * nz * nz - 1.f);
        Y[7] = -2.4270509831f * nx * nz;
        Y[8] =  1.2135254916f * (nx * nx - ny * ny);

        const float* g = sG + lane * 27;
        float fs0 = fsws[((size_t)v * 3 + 0) * 32 + lane];
        float fs1 = fsws[((size_t)v * 3 + 1) * 32 + lane];
        float fs2 = fsws[((size_t)v * 3 + 2) * 32 + lane];
        #pragma unroll
        for (int c = 0; c < 3; ++c) {
            float light = 0.f;
            #pragma unroll
            for (int k = 0; k < 9; ++k) light += Y[k] * g[c * 9 + k];
            float texc = texws[((size_t)v * 3 + c) * 32 + lane];
            float fstc = R[c * 3 + 0] * fs0 + R[c * 3 + 1] * fs1 + R[c * 3 + 2] * fs2
                       + sT[lane * 3 + c];
            sOut[(y * 32 + lane) * 6 + c]     = texc * light;
            sOut[(y * 32 + lane) * 6 + 3 + c] = fstc;
        }
    }
    __syncthreads();

    int v0 = blockIdx.x * 8;
    for (int idx = tid; idx < 32 * 24; idx += 256) {
        int b = idx / 24, t = idx % 24;
        int yy = t / 3, c = t % 3;
        int vv = v0 + yy;
        if (vv < N_VERT) {
            size_t o = (size_t)b * (N_VERT * 3) + (size_t)vv * 3 + c;
            out_color[o] = sOut[(yy * 32 + b) * 6 + c];
            out_fst[o]   = sOut[(yy * 32 + b) * 6 + 3 + c];
        }
    }
}

// ---------------------------------------------------------------------------
// Kernel D: landmark projection from fst.
// ---------------------------------------------------------------------------
__global__ __launch_bounds__(256) void k_lm(const float* __restrict__ fst,
                                            const int* __restrict__ keypoints,
                                            float* __restrict__ out_lm) {
    int i = blockIdx.x * 256 + threadIdx.x;
    if (i >= BATCH * 68) return;
    int b = i / 68, k = i % 68;
    int v = keypoints[k];
    const float* p = fst + (size_t)b * (N_VERT * 3) + (size_t)v * 3;
    float x = p[0], y = p[1], z = p[2];
    float zp = 10.f - z;                 // lm_cam.z after reverse_z + camera_pos
    float invz = 1.f / zp;
    out_lm[b * 136 + k * 2 + 0] = (1015.f * x + 112.f * zp) * invz;
    out_lm[b * 136 + k * 2 + 1] = (1015.f * y + 112.f * zp) * invz;
}

// ---------------------------------------------------------------------------
extern "C" void kernel_launch(void* const* d_in, const int* in_sizes, int n_in,
                              void* d_out, int out_size, void* d_ws, size_t ws_size,
                              hipStream_t stream) {
    (void)in_sizes; (void)n_in; (void)out_size; (void)ws_size;
    const float* coeff     = (const float*)d_in[0];
    const float* idBase    = (const float*)d_in[1];
    const float* exBase    = (const float*)d_in[2];
    const float* texBase   = (const float*)d_in[3];
    const float* meanshape = (const float*)d_in[4];
    const float* meantex   = (const float*)d_in[5];
    const int*   face_buf  = (const int*)d_in[6];
    const int*   point_buf = (const int*)d_in[7];
    const int*   keypoints = (const int*)d_in[8];

    // workspace carve-up (floats)
    float* ws     = (float*)d_ws;
    float* center = ws;                                   // 3 (+pad)
    float* Rws    = ws + 16;                              // 32*9 = 288
    float* fsws   = ws + 512;                             // M3*32
    float* texws  = fsws + (size_t)M3 * 32;               // M3*32
    float* fnws   = texws + (size_t)M3 * 32;              // (N_FACE+1)*3*32

    // output carve-up
    float* out_color = (float*)d_out;                              // B*N_VERT*3
    float* out_lm    = out_color + (size_t)BATCH * N_VERT * 3;     // B*68*2
    float* out_fst   = out_lm + (size_t)BATCH * 68 * 2;            // B*N_VERT*3

    k_prep<<<1, 256, 0, stream>>>(coeff, meanshape, center, Rws);

    k_gemm<<<(NTILES + 7) / 8, 256, 0, stream>>>(idBase, exBase, texBase, coeff,
                                                 meanshape, meantex, center, fsws, texws);

    dim3 blk(32, 8);
    k_face<<<(N_FACE + 1 + 7) / 8, blk, 0, stream>>>(fsws, face_buf, fnws);

    k_vert<<<(N_VERT + 7) / 8, blk, 0, stream>>>(fsws, texws, fnws, Rws, coeff,
                                                 point_buf, out_color, out_fst);

    k_lm<<<(BATCH * 68 + 255) / 256, 256, 0, stream>>>(out_fst, keypoints, out_lm);
}